// rcs_middle_fusion_gamma_64029372449488
// MI455X (gfx1250) — compile-verified
//
#include <hip/hip_runtime.h>
#include <math.h>

#define HW 128
#define CL 64     // lidar channels
#define CF 128    // fused channels
#define COUT 64

typedef __attribute__((ext_vector_type(2))) float v2f;
typedef __attribute__((ext_vector_type(8))) float v8f;

// ---------------------------------------------------------------- zero ws
__global__ void zero_f32_kernel(float* __restrict__ p, int n) {
    int i = blockIdx.x * blockDim.x + threadIdx.x;
    if (i < n) p[i] = 0.0f;
}

// ---------------------------------------------------------------- heatmaps
// One block per voxel; splat truncated gaussian into hm / hmf with atomic max.
// All values are >= 0 so uint bit-pattern max == float max (init is 0).
__global__ void draw_maps_kernel(const long long* __restrict__ rcoors,
                                 const float* __restrict__ rcs,
                                 const float* __restrict__ gamma,
                                 float* __restrict__ hm,
                                 float* __restrict__ hmf,
                                 int nvox) {
    int v = blockIdx.x;
    if (v >= nvox) return;
    int b = (int)rcoors[(size_t)v * 4 + 0];
    int y = (int)rcoors[(size_t)v * 4 + 2];
    int x = (int)rcoors[(size_t)v * 4 + 3];
    float tr  = fmaxf(rcs[v], 0.0f);
    float r   = floorf(gamma[0] * tr + 1.0f);
    float sig = (2.0f * r + 1.0f) * (1.0f / 6.0f);
    float inv = 1.0f / (2.0f * sig * sig);
    int   ri  = (int)r;
    int side  = 2 * ri + 1;
    int area  = side * side;
    const float eps = 1.1920929e-7f; // float32 machine eps (matches jnp.finfo eps)

    for (int idx = threadIdx.x; idx < area; idx += blockDim.x) {
        int dy = idx / side - ri;
        int dx = idx % side - ri;
        int yy = y + dy, xx = x + dx;
        if ((unsigned)yy >= HW || (unsigned)xx >= HW) continue;
        float d2 = (float)(dy * dy + dx * dx);
        float g  = expf(-d2 * inv);
        if (g < eps) continue;           // reference zeroes sub-eps values
        size_t p = ((size_t)b * HW + yy) * HW + xx;
        atomicMax((unsigned int*)&hm[p],  __float_as_uint(g));
        atomicMax((unsigned int*)&hmf[p], __float_as_uint(g * tr));
    }
}

// ---------------------------------------------------------------- weight pack
// comp_w (OIHW: [co][ci][ky][kx]) -> wp[tap][q][co][ci%4]
// so an A-fragment lane load is one contiguous 8-byte read.
__global__ void pack_weights_kernel(const float* __restrict__ compw,
                                    float* __restrict__ wp) {
    int idx = blockIdx.x * blockDim.x + threadIdx.x;
    if (idx >= 9 * 32 * 64 * 4) return;
    int r   = idx & 3;
    int co  = (idx >> 2) & 63;
    int q   = (idx >> 8) & 31;
    int tap = idx >> 13;
    int ky = tap / 3, kx = tap - ky * 3;
    int ci = 4 * q + r;
    wp[idx] = compw[((size_t)(co * CF + ci) * 3 + ky) * 3 + kx];
}

// ---------------------------------------------------------------- fused conv
// Implicit GEMM with V_WMMA_F32_16X16X4_F32:
//   D[m=co(16), n=px(16)] += A[m][k] * B[k][n],  K = 9 taps * 128 ci.
// Block = 128 threads = 4 waves; wave w handles co in [16w, 16w+16).
// Each wave computes TWO 16-pixel N-tiles, reusing every A (weight) fragment
// for two WMMAs -> halves global weight traffic per output.
__global__ __launch_bounds__(128)
void conv_wmma_kernel(const float* __restrict__ lidar,
                      const float* __restrict__ hm,
                      const float* __restrict__ hmf,
                      const float* __restrict__ attw,
                      const float* __restrict__ attb,
                      const float* __restrict__ wp,
                      const float* __restrict__ compb,
                      float* __restrict__ out) {
    // [q=ci/4][row 0..2][col 0..33][ci%4] : col-neighbor lanes hit distinct banks
    __shared__ float lds[32][3][34][4];

    const int x0 = blockIdx.x * 32;
    const int y  = blockIdx.y;
    const int b  = blockIdx.z;

    // Stage fused 128ch x 3row x 34col tile (zero-padded borders).
    // Channels >= 64 are the 1x1-conv rcs features, computed on the fly.
    for (int e = threadIdx.x; e < CF * 3 * 34; e += 128) {
        int ci  = e / 102;
        int rc  = e - ci * 102;
        int row = rc / 34;
        int col = rc - row * 34;
        int yy  = y + row - 1;
        int xx  = x0 + col - 1;
        float v = 0.0f;
        if ((unsigned)yy < HW && (unsigned)xx < HW) {
            if (ci < CL) {
                v = lidar[(((size_t)b * CL + ci) * HW + yy) * HW + xx];
            } else {
                int c = ci - CL;
                size_t p = ((size_t)b * HW + yy) * HW + xx;
                v = attw[c * 2 + 0] * hm[p] + attw[c * 2 + 1] * hmf[p] + attb[c];
            }
        }
        lds[ci >> 2][row][col][ci & 3] = v;
    }
    __syncthreads();

    const int lane    = threadIdx.x & 31;
    const int wave    = threadIdx.x >> 5;
    const int nm      = lane & 15;              // M (co) index for A, N (px) for B/D
    const int koff    = (lane < 16) ? 0 : 2;    // K pair selected by lane half
    const int co_base = wave * 16;

    v8f acc0 = {};
    v8f acc1 = {};
#pragma unroll
    for (int tap = 0; tap < 9; ++tap) {
        int ky = tap / 3, kx = tap - ky * 3;
        const float* wpt = wp + ((size_t)(tap * 32) * 64 + co_base + nm) * 4 + koff;
        const float* bpt = &lds[0][ky][nm + kx][koff];
        for (int q = 0; q < 32; ++q) {
            v2f a  = *(const v2f*)(wpt + (size_t)q * 256);      // [q][co][k%4]
            v2f b0 = *(const v2f*)(bpt + (size_t)q * 408);      // lds[q] stride 3*34*4
            v2f b1 = *(const v2f*)(bpt + (size_t)q * 408 + 64); // +16 cols
            acc0 = __builtin_amdgcn_wmma_f32_16x16x4_f32(
                false, a, false, b0, (short)0, acc0, false, false);
            acc1 = __builtin_amdgcn_wmma_f32_16x16x4_f32(
                false, a, false, b1, (short)0, acc1, false, false);
        }
    }

    const int mh = (lane < 16) ? 0 : 8;         // D layout: vgpr r -> M = r (+8)
#pragma unroll
    for (int rr = 0; rr < 8; ++rr) {
        int co = co_base + rr + mh;
        float bias = compb[co];
        size_t base = (((size_t)b * COUT + co) * HW + y) * HW + x0;
        out[base + nm]      = acc0[rr] + bias;
        out[base + 16 + nm] = acc1[rr] + bias;
    }
}

// ---------------------------------------------------------------- launch
extern "C" void kernel_launch(void* const* d_in, const int* in_sizes, int n_in,
                              void* d_out, int out_size, void* d_ws, size_t ws_size,
                              hipStream_t stream) {
    const float*     lidar  = (const float*)d_in[0];
    // d_in[1] (voxels_feat_radar) is unused by the reference
    const long long* rcoors = (const long long*)d_in[2];   // int64
    // d_in[3] (batch_size) not needed: derive B from sizes
    const float*     rcs    = (const float*)d_in[4];
    const float*     gamma  = (const float*)d_in[5];
    const float*     attw   = (const float*)d_in[6];
    const float*     attb   = (const float*)d_in[7];
    const float*     compw  = (const float*)d_in[8];
    const float*     compb  = (const float*)d_in[9];
    float*           out    = (float*)d_out;

    const int B    = in_sizes[0] / (CL * HW * HW);
    const int nvox = in_sizes[2] / 4;

    float* hm  = (float*)d_ws;
    float* hmf = hm + (size_t)B * HW * HW;
    float* wp  = hmf + (size_t)B * HW * HW;    // 9*32*64*4 floats

    const int nz = 2 * B * HW * HW;
    zero_f32_kernel<<<(nz + 255) / 256, 256, 0, stream>>>(hm, nz);
    draw_maps_kernel<<<nvox, 256, 0, stream>>>(rcoors, rcs, gamma, hm, hmf, nvox);
    pack_weights_kernel<<<(9 * 32 * 64 * 4 + 255) / 256, 256, 0, stream>>>(compw, wp);
    conv_wmma_kernel<<<dim3(HW / 32, HW, B), 128, 0, stream>>>(
        lidar, hm, hmf, attw, attb, wp, compb, out);
}